// VLLMKVCache_72155450573433
// MI455X (gfx1250) — compile-verified
//
#include <hip/hip_runtime.h>

// ---------------------------------------------------------------------------
// VLLM KV-cache scatter-copy for MI455X (gfx1250).
// Pure HBM-bandwidth problem: read cache (1 GiB) + write out (1 GiB) +
// input rows (2 x 32 MiB) ~= 2.06 GiB -> ~95 us at 23.3 TB/s. Zero FLOPs,
// so the CDNA5 paths that matter are data movement: async global<->LDS b128
// ops (ASYNCcnt) pipelined 4-deep with partial waits, plus NT b128 stores.
// ---------------------------------------------------------------------------

typedef float float4v __attribute__((ext_vector_type(4)));
typedef int   int4v   __attribute__((ext_vector_type(4)));

#define AS_GLOBAL __attribute__((address_space(1)))
#define AS_LDS    __attribute__((address_space(3)))

#if defined(__HIP_DEVICE_COMPILE__) && \
    __has_builtin(__builtin_amdgcn_global_load_async_to_lds_b128) && \
    __has_builtin(__builtin_amdgcn_global_store_async_from_lds_b128) && \
    __has_builtin(__builtin_amdgcn_s_wait_asynccnt)
#define USE_ASYNC_LDS_COPY 1
#else
#define USE_ASYNC_LDS_COPY 0
#endif

// ---------------------------------------------------------------------------
// Kernel 1: bulk copy cache -> out.
// Async path: 4-stage LDS pipeline, 16B/lane/iter.
//   Safety argument (loads complete in-order among loads, stores in-order
//   among stores, types mutually unordered):
//   * "wait asynccnt<=1" before store_k guarantees load_k is complete as long
//     as load_{k+1} is also outstanding (any 2-done combination implies it).
//     Only the final iteration lacks a younger load -> it uses "wait 0".
//   * 4 LDS stages: the load issued at iter k targets stage (k+2)&3, while
//     possibly-outstanding stores read stages (k-1)&3 and k&3, and the
//     possibly-outstanding load writes (k+1)&3 -- all distinct mod 4.
// ---------------------------------------------------------------------------
__global__ __launch_bounds__(256) void kvcache_bulk_copy(
    const float* __restrict__ src, float* __restrict__ dst,
    int iters, long long n_vec4) {
  const long long stride = (long long)gridDim.x * 256;
  const long long i0     = (long long)blockIdx.x * 256 + threadIdx.x;

#if USE_ASYNC_LDS_COPY
  __shared__ float4v lds_buf[4][256];

  // Prologue: prefetch chunks 0 and 1.
  if (iters > 0)
    __builtin_amdgcn_global_load_async_to_lds_b128(
        (AS_GLOBAL int4v*)((const float4v*)src + i0),
        (AS_LDS int4v*)&lds_buf[0][threadIdx.x], 0, 0);
  if (iters > 1)
    __builtin_amdgcn_global_load_async_to_lds_b128(
        (AS_GLOBAL int4v*)((const float4v*)src + i0 + stride),
        (AS_LDS int4v*)&lds_buf[1][threadIdx.x], 0, 0);

  for (int k = 0; k < iters; ++k) {
    if (k < iters - 1) {
      __builtin_amdgcn_s_wait_asynccnt(1);   // load_k provably complete
    } else {
      __builtin_amdgcn_s_wait_asynccnt(0);   // tail: full drain before store
    }
    __builtin_amdgcn_global_store_async_from_lds_b128(
        (AS_GLOBAL int4v*)((float4v*)dst + i0 + (long long)k * stride),
        (AS_LDS int4v*)&lds_buf[k & 3][threadIdx.x], 0, 0);
    const int k2 = k + 2;
    if (k2 < iters)
      __builtin_amdgcn_global_load_async_to_lds_b128(
          (AS_GLOBAL int4v*)((const float4v*)src + i0 + (long long)k2 * stride),
          (AS_LDS int4v*)&lds_buf[k2 & 3][threadIdx.x], 0, 0);
  }
  __builtin_amdgcn_s_wait_asynccnt(0);
#else
  for (long long i = i0; i < (long long)iters * stride; i += stride) {
    float4v v = __builtin_nontemporal_load((const float4v*)src + i);
    __builtin_nontemporal_store(v, (float4v*)dst + i);
  }
#endif

  // Remainder (grid doesn't divide n_vec4 exactly): one NT b128 per thread.
  const long long r = i0 + (long long)iters * stride;
  if (r < n_vec4) {
    float4v v = __builtin_nontemporal_load((const float4v*)src + r);
    __builtin_nontemporal_store(v, (float4v*)dst + r);
  }
}

// ---------------------------------------------------------------------------
// Kernel 2: scatter token rows into out. One block per token; each row is
// 4 KiB contiguous -> 256 lanes x one b128 NT store. The slot_mapping load is
// block-uniform, so the compiler scalarizes it to an s_load.
// ---------------------------------------------------------------------------
__global__ __launch_bounds__(256) void kvcache_scatter_rows(
    const float* __restrict__ input, const int* __restrict__ slot_mapping,
    float* __restrict__ out, int row_vec4) {
  const int t = blockIdx.x;
  const long long slot = (long long)slot_mapping[t];
  const float4v* srow = (const float4v*)input + (long long)t * row_vec4;
  float4v*       drow = (float4v*)out + slot * row_vec4;
  for (int i = threadIdx.x; i < row_vec4; i += blockDim.x) {
    float4v v = srow[i];
    __builtin_nontemporal_store(v, &drow[i]);
  }
}

// ---------------------------------------------------------------------------
// Host launcher.
//   d_in[0] = input        (NUM_TOKENS, NUM_KV_HEADS, HEAD_DIM) f32
//   d_in[1] = cache        (NUM_SLOTS,  NUM_KV_HEADS, HEAD_DIM) f32
//   d_in[2] = slot_mapping (NUM_TOKENS,) i32
//   d_out   = updated cache, same shape as d_in[1]
// Kernel order on one stream guarantees scatter overwrites after the copy.
// ---------------------------------------------------------------------------
extern "C" void kernel_launch(void* const* d_in, const int* in_sizes, int n_in,
                              void* d_out, int out_size, void* d_ws, size_t ws_size,
                              hipStream_t stream) {
  (void)n_in; (void)d_ws; (void)ws_size;

  const float* input = (const float*)d_in[0];
  const float* cache = (const float*)d_in[1];
  const int*   slots = (const int*)d_in[2];
  float*       out   = (float*)d_out;

  const long long total  = (long long)out_size;   // 268,435,456 f32 (1 GiB)
  const long long n_vec4 = total >> 2;

  const int       blocks = 8192;
  const long long stride = (long long)blocks * 256;      // 2,097,152 vec4/step
  const int       iters  = (int)(n_vec4 / stride);       // 32 full steps

  kvcache_bulk_copy<<<blocks, 256, 0, stream>>>(cache, out, iters, n_vec4);

  const int num_tokens = in_sizes[2];                  // 8192
  const int row_elems  = in_sizes[0] / num_tokens;     // 1024 (= 8 heads * 128)
  kvcache_scatter_rows<<<num_tokens, 256, 0, stream>>>(input, slots, out,
                                                       row_elems >> 2);
}